// FPQuantizer_16595753632080
// MI455X (gfx1250) — compile-verified
//
#include <hip/hip_runtime.h>

// ---------------------------------------------------------------------------
// FP4 (e2m1) group fake-quantizer for MI455X (gfx1250, wave32).
// Pass 1: global abs-max (bit-pattern u32 atomic max), 2x-unrolled grid-stride
//         b128 loads for memory-level parallelism.
// Pass 2: async global->LDS double-buffered pipeline (ASYNCcnt), per-wave
//         128-float group absmax via wave32 shuffles, branchless e2m1 round.
// Memory-bound: ~128 MiB effective HBM traffic (x stays resident in 192MB L2
// between passes) => ~6 us floor at 23.3 TB/s.
// ---------------------------------------------------------------------------

#if __has_builtin(__builtin_amdgcn_s_wait_asynccnt)
#define WAIT_ASYNCCNT(N) __builtin_amdgcn_s_wait_asynccnt(N)
#else
#define WAIT_ASYNCCNT(N) asm volatile("s_wait_asynccnt %0" ::"n"(N) : "memory")
#endif

// Async copy of 16 bytes per lane from global to LDS (ASYNCcnt-tracked).
// GVS addressing: 64-bit SGPR base + 32-bit per-lane VGPR byte offset.
__device__ __forceinline__ void async_load_b128(unsigned ldsByteAddr,
                                                const float* __restrict__ base,
                                                unsigned byteOff) {
  asm volatile("global_load_async_to_lds_b128 %0, %1, %2"
               :
               : "v"(ldsByteAddr), "v"(byteOff), "s"(base)
               : "memory");
}

__device__ __forceinline__ float wave_max32(float v) {
#pragma unroll
  for (int m = 16; m > 0; m >>= 1)
    v = fmaxf(v, __shfl_xor(v, m, 32));
  return v;
}

// ---------------------------------------------------------------------------
// e2m1 nearest-grid rounding. Thresholds are midpoints of
// {0, .5, 1, 1.5, 2, 3, 4, 6}. Reference argmin picks the LOWEST grid index
// on ties: for xs>0 ties round down in magnitude (strict >), for xs<0 ties
// round toward the more negative value (>= on magnitude).
// ---------------------------------------------------------------------------
__device__ __forceinline__ float fpq_dequant1(float xv, float inv,
                                              float scale) {
  const float xs = xv * inv;  // in [-6, 6] by construction
  const float a = fabsf(xs);
  const bool neg = xs < 0.0f;
  float q = 0.0f;
  q = (neg ? (a >= 0.25f) : (a > 0.25f)) ? 0.5f : q;
  q = (neg ? (a >= 0.75f) : (a > 0.75f)) ? 1.0f : q;
  q = (neg ? (a >= 1.25f) : (a > 1.25f)) ? 1.5f : q;
  q = (neg ? (a >= 1.75f) : (a > 1.75f)) ? 2.0f : q;
  q = (neg ? (a >= 2.50f) : (a > 2.50f)) ? 3.0f : q;
  q = (neg ? (a >= 3.50f) : (a > 3.50f)) ? 4.0f : q;
  q = (neg ? (a >= 5.00f) : (a > 5.00f)) ? 6.0f : q;
  return (neg ? -q : q) * scale;
}

// ---------------------------------------------------------------------------
// Pass 2 (placed first in .text so the disasm snippet shows the async ops).
// Tile = 1024 floats per block; each wave owns one 128-float quant group
// (lane i -> 4 contiguous floats), so the group reduction is a pure wave32
// shuffle reduction and no block barrier is needed: each wave only consumes
// LDS bytes written by its own async loads, ordered by s_wait_asynccnt.
// ---------------------------------------------------------------------------
__global__ void fpq_quantize_kernel(const float* __restrict__ x,
                                    const float* __restrict__ clipStrength,
                                    const unsigned* __restrict__ gmaxBits,
                                    float* __restrict__ out, int numTiles) {
  __shared__ __align__(16) float lbuf[2][1024];  // 8 KB double buffer
  const int tid = threadIdx.x;
  const float clip = clipStrength[0] * __uint_as_float(gmaxBits[0]);

  unsigned ldsAddr[2];
  ldsAddr[0] = (unsigned)(unsigned long long)&lbuf[0][tid * 4];
  ldsAddr[1] = (unsigned)(unsigned long long)&lbuf[1][tid * 4];

  const int stride = gridDim.x;
  int t = blockIdx.x;
  int stage = 0;

  if (t < numTiles)
    async_load_b128(ldsAddr[0], x,
                    ((unsigned)t * 1024u + (unsigned)tid * 4u) * 4u);

  for (; t < numTiles; t += stride) {
    const int nxt = t + stride;
    if (nxt < numTiles) {  // uniform branch: prefetch next tile, keep 1 inflight
      async_load_b128(ldsAddr[stage ^ 1], x,
                      ((unsigned)nxt * 1024u + (unsigned)tid * 4u) * 4u);
      WAIT_ASYNCCNT(1);
    } else {
      WAIT_ASYNCCNT(0);
    }

    float4 v = *(const float4*)&lbuf[stage][tid * 4];

    // Global clip (reference clips before the group absmax).
    v.x = fminf(fmaxf(v.x, -clip), clip);
    v.y = fminf(fmaxf(v.y, -clip), clip);
    v.z = fminf(fmaxf(v.z, -clip), clip);
    v.w = fminf(fmaxf(v.w, -clip), clip);

    // Group (128-elem) absmax across this wave.
    float lm = fmaxf(fmaxf(fabsf(v.x), fabsf(v.y)),
                     fmaxf(fabsf(v.z), fabsf(v.w)));
    const float gm = wave_max32(lm);
    const float scale = gm * (1.0f / 6.0f);
    const float inv = gm > 0.0f ? 6.0f / gm : 0.0f;

    float4 r;
    r.x = fpq_dequant1(v.x, inv, scale);
    r.y = fpq_dequant1(v.y, inv, scale);
    r.z = fpq_dequant1(v.z, inv, scale);
    r.w = fpq_dequant1(v.w, inv, scale);

    *(float4*)(out + (size_t)t * 1024 + (size_t)tid * 4) = r;
    stage ^= 1;
  }
}

// ---------------------------------------------------------------------------
// Pass 1: global max|x| reduction. 2x unrolled grid-stride so two b128 loads
// are in flight per wave before the first consumption wait.
// ---------------------------------------------------------------------------
__global__ void fpq_absmax_kernel(const float* __restrict__ x,
                                  unsigned* __restrict__ gmaxBits,
                                  long long nvec) {
  __shared__ float smax[8];
  const float4* __restrict__ xv = (const float4*)x;
  long long i = (long long)blockIdx.x * blockDim.x + threadIdx.x;
  const long long stride = (long long)gridDim.x * blockDim.x;
  float m = 0.0f;
  // 2x unrolled: two independent loads in flight.
  for (; i + stride < nvec; i += 2 * stride) {
    float4 a = xv[i];
    float4 b = xv[i + stride];
    float ma = fmaxf(fmaxf(fabsf(a.x), fabsf(a.y)),
                     fmaxf(fabsf(a.z), fabsf(a.w)));
    float mb = fmaxf(fmaxf(fabsf(b.x), fabsf(b.y)),
                     fmaxf(fabsf(b.z), fabsf(b.w)));
    m = fmaxf(m, fmaxf(ma, mb));
  }
  if (i < nvec) {
    float4 a = xv[i];
    m = fmaxf(m, fmaxf(fmaxf(fabsf(a.x), fabsf(a.y)),
                       fmaxf(fabsf(a.z), fabsf(a.w))));
  }
  m = wave_max32(m);
  const int wid = threadIdx.x >> 5;
  const int lid = threadIdx.x & 31;
  if (lid == 0) smax[wid] = m;
  __syncthreads();
  if (threadIdx.x == 0) {
    float bm = smax[0];
    const int nw = blockDim.x >> 5;
#pragma unroll
    for (int w = 1; w < 8; ++w)
      if (w < nw) bm = fmaxf(bm, smax[w]);
    // |x| >= 0, so IEEE bit pattern order == float order: u32 atomic max.
    atomicMax(gmaxBits, __float_as_uint(bm));
  }
}

// ---------------------------------------------------------------------------
// Host launcher. d_in[0] = x (f32, 4096*4096), d_in[1] = clipping_strength
// (f32 scalar). d_out = f32, same shape as x. d_ws[0..3] = absmax bits.
// ---------------------------------------------------------------------------
extern "C" void kernel_launch(void* const* d_in, const int* in_sizes, int n_in,
                              void* d_out, int out_size, void* d_ws,
                              size_t ws_size, hipStream_t stream) {
  const float* x = (const float*)d_in[0];
  const float* cs = (const float*)d_in[1];
  float* out = (float*)d_out;
  unsigned* gmaxBits = (unsigned*)d_ws;

  const long long n = (long long)in_sizes[0];
  const long long nvec = n >> 2;        // float4 count (n divisible by 4)
  const int numTiles = (int)(n >> 10);  // 1024-float tiles (n divisible by 1024)

  hipMemsetAsync(d_ws, 0, sizeof(unsigned), stream);
  fpq_absmax_kernel<<<2048, 256, 0, stream>>>(x, gmaxBits, nvec);
  fpq_quantize_kernel<<<2048, 256, 0, stream>>>(x, cs, gmaxBits, out, numTiles);
}